// JointNetwork_72189810311722
// MI455X (gfx1250) — compile-verified
//
#include <hip/hip_runtime.h>
#include <hip/hip_bf16.h>

#define B_   8
#define T_   200
#define U_   50
#define U1_  51
#define H_   320
#define V_   1024
#define NEGF (-1.0e30f)

typedef __attribute__((ext_vector_type(2))) float v2f;
typedef __attribute__((ext_vector_type(8))) float v8f;

__device__ __forceinline__ float logaddexpf_(float x, float y) {
  float m = fmaxf(x, y);
  float d = fabsf(x - y);
  return m + log1pf(__expf(-d));
}

// C[M,N] = A[M,K] @ W[K,N] (+ bias[N]); all row-major fp32.
// One wave (32 threads) per block computes a 16x64 tile of C with
// V_WMMA_F32_16X16X4_F32, K-loop in steps of 4, A-fragment reused x4.
__global__ __launch_bounds__(32)
void wmma_gemm_f32(const float* __restrict__ A, const float* __restrict__ W,
                   const float* __restrict__ bias, float* __restrict__ C,
                   int M, int N, int K) {
  const int lane = threadIdx.x;
  const int mi = lane & 15;              // M (for A) / N (for B,C) position within tile
  const int kh = (lane >> 4) << 1;       // K pair: 0 or 2
  const int m0 = blockIdx.x * 16;
  const int n0 = blockIdx.y * 64;
  int m = m0 + mi;
  if (m >= M) m = M - 1;                 // clamp OOB rows; their results are never stored
  const float* arow = A + (size_t)m * K;
  const int n = n0 + mi;

  v8f acc0 = {}, acc1 = {}, acc2 = {}, acc3 = {};
  for (int k = 0; k < K; k += 4) {
    v2f a;
    a.x = arow[k + kh];
    a.y = arow[k + kh + 1];
    const float* w0 = W + (size_t)(k + kh) * N + n;
    const float* w1 = w0 + N;
    v2f b0, b1, b2, b3;
    b0.x = w0[0];   b0.y = w1[0];
    b1.x = w0[16];  b1.y = w1[16];
    b2.x = w0[32];  b2.y = w1[32];
    b3.x = w0[48];  b3.y = w1[48];
    acc0 = __builtin_amdgcn_wmma_f32_16x16x4_f32(false, a, false, b0, (short)0, acc0, false, false);
    acc1 = __builtin_amdgcn_wmma_f32_16x16x4_f32(false, a, false, b1, (short)0, acc1, false, false);
    acc2 = __builtin_amdgcn_wmma_f32_16x16x4_f32(false, a, false, b2, (short)0, acc2, false, false);
    acc3 = __builtin_amdgcn_wmma_f32_16x16x4_f32(false, a, false, b3, (short)0, acc3, false, false);
  }

  float bv0 = 0.f, bv1 = 0.f, bv2 = 0.f, bv3 = 0.f;
  if (bias) { bv0 = bias[n]; bv1 = bias[n + 16]; bv2 = bias[n + 32]; bv3 = bias[n + 48]; }
  const int rbase = m0 + (lane >> 4) * 8;   // C/D: VGPR i -> row i (lanes 0-15) / row i+8 (lanes 16-31)
  #pragma unroll
  for (int i = 0; i < 8; ++i) {
    int row = rbase + i;
    if (row < M) {
      float* crow = C + (size_t)row * N + n;
      crow[0]  = acc0[i] + bv0;
      crow[16] = acc1[i] + bv1;
      crow[32] = acc2[i] + bv2;
      crow[48] = acc3[i] + bv3;
    }
  }
}

// One block per (b,t,u) row. Writes logits (non-temporal, streamed 334MB),
// computes row logsumexp, and emits lp_blank / lp_label for the loss lattice.
__global__ __launch_bounds__(256)
void logits_lse_kernel(const float* __restrict__ E, const float* __restrict__ P,
                       const int* __restrict__ ys,
                       float* __restrict__ logits, float* __restrict__ lpb,
                       float* __restrict__ lpl) {
  const int r  = blockIdx.x;           // r = (b*T + t)*U1 + u
  const int u  = r % U1_;
  const int bt = r / U1_;
  const int t  = bt % T_;
  const int b  = bt / T_;
  const float* e = E + (size_t)bt * V_;
  const float* p = P + (size_t)(b * U1_ + u) * V_;   // b_ff already folded into P
  float* out = logits + (size_t)r * V_;
  const int tid = threadIdx.x;

  float x[4];
  float m = NEGF;
  #pragma unroll
  for (int k = 0; k < 4; ++k) {
    int v = tid + k * 256;
    float val = e[v] + p[v];
    x[k] = val;
    __builtin_nontemporal_store(val, out + v);
    m = fmaxf(m, val);
  }

  __shared__ float red[256];
  __shared__ float s_m, s_xb, s_xl;
  red[tid] = m;
  __syncthreads();
  #pragma unroll
  for (int s = 128; s > 0; s >>= 1) {
    if (tid < s) red[tid] = fmaxf(red[tid], red[tid + s]);
    __syncthreads();
  }
  if (tid == 0) { s_m = red[0]; s_xb = x[0]; }       // x at v=0 lives in thread 0
  if (u < U_) {
    int lab = ys[b * U_ + u];                        // in [1, V)
    if ((lab & 255) == tid) s_xl = x[lab >> 8];      // owner thread captures label logit
  }
  __syncthreads();

  const float mm = s_m;
  float sl = 0.f;
  #pragma unroll
  for (int k = 0; k < 4; ++k) sl += __expf(x[k] - mm);
  red[tid] = sl;
  __syncthreads();
  #pragma unroll
  for (int s = 128; s > 0; s >>= 1) {
    if (tid < s) red[tid] += red[tid + s];
    __syncthreads();
  }
  if (tid == 0) {
    float Z = mm + logf(red[0]);
    lpb[r] = s_xb - Z;
    if (u < U_) lpl[(size_t)(b * T_ + t) * U_ + u] = s_xl - Z;
  }
}

// Anti-diagonal wavefront over the T x U1 lattice; one block per batch element.
// alpha[t,u] = logaddexp(alpha[t-1,u] + lpb[t-1,u], alpha[t,u-1] + lpl[t,u-1])
// t=0 row: exact cumsum (left-only add); u=0 column: up-only (emit = -inf).
__global__ __launch_bounds__(64)
void alpha_kernel(const float* __restrict__ lpb, const float* __restrict__ lpl,
                  const int* __restrict__ ylen, float* __restrict__ losses) {
  const int b = blockIdx.x;
  const int u = threadIdx.x;
  __shared__ float Acur[U1_];
  __shared__ float last[U1_];

  for (int d = 0; d < T_ + U1_ - 1; ++d) {
    int t = d - u;
    bool active = (u < U1_) && (t >= 0) && (t < T_);
    float val = 0.f;
    if (active) {
      if (d == 0) {
        val = 0.f;
      } else if (t == 0) {
        val = Acur[u - 1] + lpl[(size_t)(b * T_) * U_ + (u - 1)];
      } else if (u == 0) {
        val = Acur[0] + lpb[(size_t)(b * T_ + (t - 1)) * U1_];
      } else {
        float up = Acur[u]     + lpb[(size_t)(b * T_ + (t - 1)) * U1_ + u];
        float lf = Acur[u - 1] + lpl[(size_t)(b * T_ + t) * U_ + (u - 1)];
        val = logaddexpf_(up, lf);
      }
    }
    __syncthreads();                    // all reads of Acur done
    if (active) {
      Acur[u] = val;
      if (t == T_ - 1) last[u] = val;
    }
    __syncthreads();                    // writes visible for next diagonal
  }

  if (u == 0) {
    int yl = ylen[b];
    float ae = last[yl];
    float be = lpb[(size_t)(b * T_ + (T_ - 1)) * U1_ + yl];
    losses[b] = -(ae + be);
  }
}

__global__ void finalize_kernel(const float* __restrict__ losses, float* __restrict__ out) {
  if (threadIdx.x == 0) {
    float s = 0.f;
    for (int i = 0; i < B_; ++i) s += losses[i];
    out[0] = s / (float)B_;
  }
}

extern "C" void kernel_launch(void* const* d_in, const int* in_sizes, int n_in,
                              void* d_out, int out_size, void* d_ws, size_t ws_size,
                              hipStream_t stream) {
  const float* enc    = (const float*)d_in[0];   // (B,T,H)
  const float* pred   = (const float*)d_in[1];   // (B,U1,H)
  const int*   ys     = (const int*)d_in[4];     // (B,U)
  const int*   ylen   = (const int*)d_in[5];     // (B,)
  const float* W_enc  = (const float*)d_in[6];
  const float* b_enc  = (const float*)d_in[7];
  const float* W_pred = (const float*)d_in[8];
  const float* b_pred = (const float*)d_in[9];
  const float* W_ffe  = (const float*)d_in[10];
  const float* W_ffp  = (const float*)d_in[11];
  const float* b_ff   = (const float*)d_in[12];

  float* ws = (float*)d_ws;
  float* enc_proj  = ws;                               // 1600*320
  float* pred_proj = enc_proj  + 1600 * 320;           // 408*320
  float* E         = pred_proj + 408 * 320;            // 1600*1024
  float* P         = E         + 1600 * 1024;          // 408*1024
  float* lpb       = P         + 408 * 1024;           // B*T*U1
  float* lpl       = lpb       + B_ * T_ * U1_;        // B*T*U
  float* losses    = lpl       + B_ * T_ * U_;         // B

  float* loss_out = (float*)d_out;
  float* logits   = loss_out + 1;                      // tuple: (loss, logits)

  // enc_proj = enc @ W_enc + b_enc          (1600x320 = 1600x320 @ 320x320)
  wmma_gemm_f32<<<dim3(100, 5), 32, 0, stream>>>(enc, W_enc, b_enc, enc_proj, 1600, 320, 320);
  // pred_proj = pred @ W_pred + b_pred      (408x320)
  wmma_gemm_f32<<<dim3(26, 5), 32, 0, stream>>>(pred, W_pred, b_pred, pred_proj, 408, 320, 320);
  // E = enc_proj @ W_ff_enc                 (1600x1024)
  wmma_gemm_f32<<<dim3(100, 16), 32, 0, stream>>>(enc_proj, W_ffe, nullptr, E, 1600, 1024, 320);
  // P = pred_proj @ W_ff_pred + b_ff        (408x1024), b_ff folded here
  wmma_gemm_f32<<<dim3(26, 16), 32, 0, stream>>>(pred_proj, W_ffp, b_ff, P, 408, 1024, 320);
  // logits (streamed) + per-row logsumexp + blank/label log-probs
  logits_lse_kernel<<<dim3(B_ * T_ * U1_), 256, 0, stream>>>(E, P, ys, logits, lpb, lpl);
  // wavefront alpha recursion
  alpha_kernel<<<dim3(B_), 64, 0, stream>>>(lpb, lpl, ylen, losses);
  // mean over batch -> d_out[0]
  finalize_kernel<<<1, 32, 0, stream>>>(losses, loss_out);
}